// SVMModel_45792941310118
// MI455X (gfx1250) — compile-verified
//
#include <hip/hip_runtime.h>
#include <hip/hip_bf16.h>

typedef __attribute__((ext_vector_type(2))) float v2f;
typedef __attribute__((ext_vector_type(8))) float v8f;

#define BATCH   32768
#define NSV     8192
#define DIM     32
#define NELEM   (BATCH * DIM)   // 1048576

#if __has_builtin(__builtin_amdgcn_exp2f)
#define EXP2(x) __builtin_amdgcn_exp2f(x)
#else
#define EXP2(x) exp2f(x)
#endif

// ---------------- Pass 1: partial sums for var(x) (deterministic) ----------------
__global__ __launch_bounds__(256) void svm_reduce_var(const float* __restrict__ x,
                                                      float* __restrict__ psum,
                                                      float* __restrict__ psumsq) {
    __shared__ float ssum[256];
    __shared__ float ssq[256];
    const int tid = threadIdx.x;
    const int base = blockIdx.x * 1024 + tid;   // 1024 blocks * 1024 elems
    float s = 0.f, q = 0.f;
#pragma unroll
    for (int k = 0; k < 4; ++k) {
        float v = x[base + k * 256];
        s += v;
        q += v * v;
    }
    ssum[tid] = s; ssq[tid] = q;
    __syncthreads();
    for (int off = 128; off > 0; off >>= 1) {
        if (tid < off) { ssum[tid] += ssum[tid + off]; ssq[tid] += ssq[tid + off]; }
        __syncthreads();
    }
    if (tid == 0) { psum[blockIdx.x] = ssum[0]; psumsq[blockIdx.x] = ssq[0]; }
}

// ---------------- Pass 2: finalize gamma = 1 / (d * var(x)) ----------------
__global__ __launch_bounds__(256) void svm_finalize_gamma(const float* __restrict__ psum,
                                                          const float* __restrict__ psumsq,
                                                          float* __restrict__ gamma_out) {
    __shared__ float ssum[256];
    __shared__ float ssq[256];
    const int tid = threadIdx.x;
    float s = 0.f, q = 0.f;
#pragma unroll
    for (int k = 0; k < 4; ++k) { s += psum[tid + k * 256]; q += psumsq[tid + k * 256]; }
    ssum[tid] = s; ssq[tid] = q;
    __syncthreads();
    for (int off = 128; off > 0; off >>= 1) {
        if (tid < off) { ssum[tid] += ssum[tid + off]; ssq[tid] += ssq[tid + off]; }
        __syncthreads();
    }
    if (tid == 0) {
        const float n = (float)NELEM;
        float mean = ssum[0] / n;
        float var  = ssq[0] / n - mean * mean;
        gamma_out[0] = 1.0f / ((float)DIM * var);
    }
}

// ---------------- Prep: sv row norms and w = alpha * label ----------------
__global__ __launch_bounds__(256) void svm_sv_prep(const float* __restrict__ sv,
                                                   const float* __restrict__ labels,
                                                   const float* __restrict__ alphas,
                                                   float* __restrict__ svnorm,
                                                   float* __restrict__ w) {
    const int j = blockIdx.x * blockDim.x + threadIdx.x;   // 0..8191
    const float* row = sv + j * DIM;
    float s = 0.f;
#pragma unroll
    for (int k = 0; k < DIM; ++k) s = fmaf(row[k], row[k], s);
    svnorm[j] = s;
    w[j] = alphas[j] * labels[j];
}

// ---------------- Main kernel helpers ----------------
__device__ __forceinline__ void load_btile(const float* __restrict__ sb, v2f bm[8]) {
#pragma unroll
    for (int i = 0; i < 8; ++i) bm[i] = *(const v2f*)(sb + 4 * i);
}

__device__ __forceinline__ void compute_tile(const v2f a[8], const v2f bm[8],
                                             float svn, float wv,
                                             const float* xg, float tg2, float g2,
                                             float acc[8]) {
    v8f c = 0.0f;
#pragma unroll
    for (int i = 0; i < 8; ++i) {
        // D = A*B + C ; f32 in / f32 accumulate (exact-precision GEMM part)
        c = __builtin_amdgcn_wmma_f32_16x16x4_f32(false, a[i], false, bm[i],
                                                  (short)0, c, false, false);
    }
    const float sg = -g2 * svn;
#pragma unroll
    for (int r = 0; r < 8; ++r) {
        // exp(-gamma*(xn + svn - 2S)) = exp2(xg[r] + sg + tg2*S)
        float t = fmaf(tg2, c[r], xg[r] + sg);
        acc[r] = fmaf(wv, EXP2(t), acc[r]);
    }
}

// ---------------- Main: WMMA f32 16x16x4 GEMM + RBF + weighted reduction ----------------
__global__ __launch_bounds__(128) void svm_main(const float* __restrict__ x,
                                                const float* __restrict__ sv,
                                                const float* __restrict__ svnorm,
                                                const float* __restrict__ w,
                                                const float* __restrict__ gamma_p,
                                                const float* __restrict__ bias,
                                                float* __restrict__ out) {
    const int lane  = threadIdx.x & 31;
    const int h     = lane >> 4;      // 0: K-low half, 1: K-high half
    const int lidx  = lane & 15;      // column / row-in-tile index
    const int wave  = (blockIdx.x * blockDim.x + threadIdx.x) >> 5;
    const int m_base = wave * 16;

    const float gamma = gamma_p[0];
    const float g2  = gamma * 1.4426950408889634f;   // gamma * log2(e)
    const float tg2 = 2.0f * g2;

    // ---- A tile: x[m_base .. m_base+15][0..31] in WMMA f32 16x16x4 A layout.
    // Lane L holds A-slice i element (v): x[m_base + L%16][4*i + 2*(L/16) + v]
    v2f a[8];
    const float* xrow = x + (m_base + lidx) * DIM + 2 * h;
#pragma unroll
    for (int i = 0; i < 8; ++i) a[i] = *(const v2f*)(xrow + 4 * i);

    // ---- Row norms from the A registers (partner lane L^16 has the other half).
    float p = 0.f;
#pragma unroll
    for (int i = 0; i < 8; ++i) p = fmaf(a[i].x, a[i].x, fmaf(a[i].y, a[i].y, p));
    p += __shfl_xor(p, 16, 32);                 // ||x[m_base + lidx]||^2
    float xg[8];
#pragma unroll
    for (int r = 0; r < 8; ++r) xg[r] = -g2 * __shfl(p, 8 * h + r, 32);

    float acc[8];
#pragma unroll
    for (int r = 0; r < 8; ++r) acc[r] = 0.f;

    // Per-lane B base: B-slice i element (v) = sv[nb + L%16][4*i + 2*(L/16) + v]
    const float* svbase = sv + lidx * DIM + 2 * h;

    // ---- Double-buffered pipeline over 512 N-tiles (2 tiles per iteration).
    v2f bmA[8], bmB[8];
    float svnA, wvA, svnB, wvB;
    load_btile(svbase, bmA);
    svnA = svnorm[lidx];
    wvA  = w[lidx];

    for (int nb = 0; nb < NSV; nb += 32) {
        // prefetch tile nb+16 into B buffer
        load_btile(svbase + (nb + 16) * DIM, bmB);
        svnB = svnorm[nb + 16 + lidx];
        wvB  = w[nb + 16 + lidx];

        compute_tile(a, bmA, svnA, wvA, xg, tg2, g2, acc);

        // prefetch tile nb+32 into A buffer (wraps harmlessly to 0 on last iter)
        const int nb2 = (nb + 32) & (NSV - 1);
        load_btile(svbase + nb2 * DIM, bmA);
        svnA = svnorm[nb2 + lidx];
        wvA  = w[nb2 + lidx];

        compute_tile(a, bmB, svnB, wvB, xg, tg2, g2, acc);
    }

    // ---- Reduce the 16 column-partials within each half-wave (bit 4 preserved).
#pragma unroll
    for (int r = 0; r < 8; ++r) {
        acc[r] += __shfl_xor(acc[r], 1, 32);
        acc[r] += __shfl_xor(acc[r], 2, 32);
        acc[r] += __shfl_xor(acc[r], 4, 32);
        acc[r] += __shfl_xor(acc[r], 8, 32);
    }

    if (lidx == 0) {
        const float bb = bias[0];
#pragma unroll
        for (int r = 0; r < 8; ++r) out[m_base + 8 * h + r] = acc[r] + bb;
    }
}

extern "C" void kernel_launch(void* const* d_in, const int* in_sizes, int n_in,
                              void* d_out, int out_size, void* d_ws, size_t ws_size,
                              hipStream_t stream) {
    const float* x      = (const float*)d_in[0];
    const float* sv     = (const float*)d_in[1];
    const float* labels = (const float*)d_in[2];
    const float* alphas = (const float*)d_in[3];
    const float* bias   = (const float*)d_in[4];
    float* out = (float*)d_out;

    float* ws     = (float*)d_ws;
    float* gamma  = ws;                  // 1 float
    float* psum   = ws + 16;             // 1024 floats
    float* psumsq = ws + 16 + 1024;      // 1024 floats
    float* svnorm = ws + 4096;           // 8192 floats
    float* wvec   = ws + 4096 + NSV;     // 8192 floats  (~80 KB total)

    svm_reduce_var<<<1024, 256, 0, stream>>>(x, psum, psumsq);
    svm_finalize_gamma<<<1, 256, 0, stream>>>(psum, psumsq, gamma);
    svm_sv_prep<<<NSV / 256, 256, 0, stream>>>(sv, labels, alphas, svnorm, wvec);
    // 512 blocks * 128 threads = 2048 waves * 16 rows = 32768 batch rows
    svm_main<<<BATCH / (4 * 16), 128, 0, stream>>>(x, sv, svnorm, wvec, gamma, bias, out);
}